// Renderer_62388694941742
// MI455X (gfx1250) — compile-verified
//
#include <hip/hip_runtime.h>
#include <hip/hip_bf16.h>

#define IMG 256
#define NEARP 0.1f
#define FARP 100.0f

typedef __attribute__((ext_vector_type(2))) float v2f;
typedef __attribute__((ext_vector_type(8))) float v8f;

// ---------------------------------------------------------------- fill
__global__ void fill_far_kernel(float* __restrict__ out, int n) {
    int i = blockIdx.x * blockDim.x + threadIdx.x;
    if (i < n) out[i] = FARP;
}

// ---------------------------------------------------------------- projection
__global__ void project_kernel(const float* __restrict__ verts,
                               const float* __restrict__ K,
                               const float* __restrict__ R,
                               const float* __restrict__ t,
                               const int*   __restrict__ orig,
                               float* __restrict__ xs,
                               float* __restrict__ ys,
                               float* __restrict__ zs,
                               int Nv) {
    int n = blockIdx.x * blockDim.x + threadIdx.x;
    if (n >= Nv) return;
    float x = verts[n * 3 + 0], y = verts[n * 3 + 1], z = verts[n * 3 + 2];
    float vx = R[0] * x + R[1] * y + R[2] * z + t[0];
    float vy = R[3] * x + R[4] * y + R[5] * z + t[1];
    float vz = R[6] * x + R[7] * y + R[8] * z + t[2];
    float fx = K[0], cx = K[2], fy = K[4], cy = K[5];
    float os = (float)orig[0];
    float u = fx * vx + cx;
    float w = fy * vy + cy;
    xs[n] = 2.0f * u / os - 1.0f;
    ys[n] = -(2.0f * w / os - 1.0f);
    zs[n] = vz;
}

// ---------------------------------------------------------------- face setup
// Edge functions are affine in pixel coords:
//   w0 = (y1-y2)px + (x2-x1)py + (x1*y2 - x2*y1)   (and cyclic for w1, w2)
//   znum = w0*z0 + w1*z1 + w2*z2                   (affine too)
// Per-face layout (8 float2 = 64B):
//   [0..3]  A-halves : {A0,B0} {A1,B1} {A2,B2} {Az,Bz}   (read by lanes 0-15)
//   [4..7]  C-halves : {C0,0}  {C1,0}  {C2,0}  {Cz,0}    (read by lanes 16-31)
// -> each lane half reads its 32 bytes with two global_load_b128.
// Padded faces get all-zero coefficients -> area==0 -> rejected.
__global__ void face_setup_kernel(const int* __restrict__ faces,
                                  const float* __restrict__ xs,
                                  const float* __restrict__ ys,
                                  const float* __restrict__ zs,
                                  float2* __restrict__ coef,
                                  int Nf, int Fpad) {
    int f = blockIdx.x * blockDim.x + threadIdx.x;
    if (f >= Fpad) return;
    float2* c = coef + (size_t)f * 8;
    if (f >= Nf) {
        float2 zz; zz.x = 0.0f; zz.y = 0.0f;
#pragma unroll
        for (int k = 0; k < 8; ++k) c[k] = zz;
        return;
    }
    int i0 = faces[f * 3 + 0], i1 = faces[f * 3 + 1], i2 = faces[f * 3 + 2];
    float x0 = xs[i0], y0 = ys[i0], z0 = zs[i0];
    float x1 = xs[i1], y1 = ys[i1], z1 = zs[i1];
    float x2 = xs[i2], y2 = ys[i2], z2 = zs[i2];

    float A0 = y1 - y2, B0 = x2 - x1, C0 = x1 * y2 - x2 * y1;
    float A1 = y2 - y0, B1 = x0 - x2, C1 = x2 * y0 - x0 * y2;
    float A2 = y0 - y1, B2 = x1 - x0, C2 = x0 * y1 - x1 * y0;
    float Az = A0 * z0 + A1 * z1 + A2 * z2;
    float Bz = B0 * z0 + B1 * z1 + B2 * z2;
    float Cz = C0 * z0 + C1 * z1 + C2 * z2;

    float2 v;
    v.x = A0; v.y = B0; c[0] = v;
    v.x = A1; v.y = B1; c[1] = v;
    v.x = A2; v.y = B2; c[2] = v;
    v.x = Az; v.y = Bz; c[3] = v;
    v.x = C0; v.y = 0.f; c[4] = v;
    v.x = C1; v.y = 0.f; c[5] = v;
    v.x = C2; v.y = 0.f; c[6] = v;
    v.x = Cz; v.y = 0.f; c[7] = v;
}

// ---------------------------------------------------------------- raster
// One wave per (16x16 pixel tile, face slice). Per 16-face batch: four 16x4
// f32 A-matrices (w0,w1,w2,znum coefficients), then 16 pixel rows evaluated
// with V_WMMA_F32_16X16X4_F32 each (B = [px,py,1,0] per pixel).
// Inside test uses the sign-free equivalence:
//   inside  <=>  (min3(w) >= 0  ||  max3(w) <= 0)  &&  |area| > 1e-10
// which matches the reference's (w_i * sign(area) >= 0) test exactly.
__global__ void __launch_bounds__(32)
raster_kernel(const float2* __restrict__ coef,
              float* __restrict__ depth,
              int nBatch) {
    const int lane = threadIdx.x;
    const int hi   = lane >> 4;        // 0: A/B half, 1: C/0 half
    const int ln   = lane & 15;
    const int tx   = blockIdx.x & 15;
    const int ty   = blockIdx.x >> 4;
    const float inv = 1.0f / (float)IMG;

    // B-matrix per-lane constants: lanes 0-15 -> (px, py); lanes 16-31 -> (1, 0)
    const float bx = hi ? 1.0f : (2.0f * ((float)(tx * 16 + ln) + 0.5f) * inv - 1.0f);

    float rmin[16];
#pragma unroll
    for (int s = 0; s < 16; ++s) rmin[s] = FARP;

    const int stride = gridDim.y;
    for (int b = blockIdx.y; b < nBatch; b += stride) {
        // this lane's 32 contiguous bytes of coefficients for its half
        const float4* cf =
            (const float4*)(coef + ((size_t)(b * 16 + ln)) * 8 + (size_t)hi * 4);
        // prefetch batch b + stride/2 at this lane's offset (global_prefetch_b8)
        __builtin_prefetch((const void*)(cf + (size_t)stride * 32), 0, 1);

        float4 p0 = cf[0];
        float4 p1 = cf[1];
        v2f A0; A0.x = p0.x; A0.y = p0.y;
        v2f A1; A1.x = p0.z; A1.y = p0.w;
        v2f A2; A2.x = p1.x; A2.y = p1.y;
        v2f A3; A3.x = p1.z; A3.y = p1.w;

#pragma unroll
        for (int s = 0; s < 16; ++s) {
            const float by = hi ? 0.0f
                                : -(2.0f * ((float)(ty * 16 + s) + 0.5f) * inv - 1.0f);
            v2f Bm; Bm.x = bx; Bm.y = by;
            v8f zero = {};
            v8f d0 = __builtin_amdgcn_wmma_f32_16x16x4_f32(false, A0, false, Bm,
                                                           (short)0, zero, false, false);
            v8f d1 = __builtin_amdgcn_wmma_f32_16x16x4_f32(false, A1, false, Bm,
                                                           (short)0, zero, false, false);
            v8f d2 = __builtin_amdgcn_wmma_f32_16x16x4_f32(false, A2, false, Bm,
                                                           (short)0, zero, false, false);
            v8f d3 = __builtin_amdgcn_wmma_f32_16x16x4_f32(false, A3, false, Bm,
                                                           (short)0, zero, false, false);
            float m = FARP;
#pragma unroll
            for (int k = 0; k < 8; ++k) {
                float w0 = d0[k], w1 = d1[k], w2 = d2[k], zn = d3[k];
                float area = w0 + w1 + w2;                    // as in reference
                float mn = fminf(fminf(w0, w1), w2);          // v_min3_f32
                float mx = fmaxf(fmaxf(w0, w1), w2);          // v_max3_f32
                bool inside = ((mn >= 0.0f) | (mx <= 0.0f)) & (fabsf(area) > 1e-10f);
                // rcp(0) -> inf/NaN makes the range compares fail, and such
                // elements are already excluded by `inside`, so no safe-area
                // select is needed.
                float zp = zn * __builtin_amdgcn_rcpf(area);
                bool valid = inside & (zp > NEARP) & (zp < FARP);
                m = fminf(m, valid ? zp : FARP);
            }
            rmin[s] = fminf(rmin[s], m);
        }
    }

    // merge faces 0-7 (lanes 0-15) with faces 8-15 (lanes 16-31), then combine
    // across face slices with a float-as-uint atomic min (all candidates are
    // positive floats, so uint order == float order).
#pragma unroll
    for (int s = 0; s < 16; ++s) {
        float other = __shfl_xor(rmin[s], 16, 32);
        float m = fminf(rmin[s], other);
        if (lane < 16) {
            int row = ty * 16 + s;
            int col = tx * 16 + lane;
            atomicMin((unsigned int*)&depth[row * IMG + col], __float_as_uint(m));
        }
    }
}

// ---------------------------------------------------------------- launch
extern "C" void kernel_launch(void* const* d_in, const int* in_sizes, int n_in,
                              void* d_out, int out_size, void* d_ws, size_t ws_size,
                              hipStream_t stream) {
    const float* verts = (const float*)d_in[0];
    const int*   faces = (const int*)d_in[1];
    const float* Km    = (const float*)d_in[2];
    const float* Rm    = (const float*)d_in[3];
    const float* tv    = (const float*)d_in[4];
    const int*   orig  = (const int*)d_in[5];
    float* depth = (float*)d_out;

    const int Nv = in_sizes[0] / 3;
    const int Nf = in_sizes[1] / 3;   // reversed-winding duplicates are provably
                                      // identical in depth -> process Nf only
    const int Fpad   = ((Nf + 15) / 16) * 16;
    const int nBatch = Fpad / 16;

    // workspace layout: xs|ys|zs (Nv floats each), then coef (Fpad * 8 float2)
    float* xs = (float*)d_ws;
    float* ys = xs + Nv;
    float* zs = ys + Nv;
    size_t off = ((size_t)3 * (size_t)Nv * sizeof(float) + 15) & ~(size_t)15;
    float2* coef = (float2*)((char*)d_ws + off);

    const int npix = IMG * IMG;
    fill_far_kernel<<<(npix + 255) / 256, 256, 0, stream>>>(depth, npix);
    project_kernel<<<(Nv + 255) / 256, 256, 0, stream>>>(verts, Km, Rm, tv, orig,
                                                         xs, ys, zs, Nv);
    face_setup_kernel<<<(Fpad + 255) / 256, 256, 0, stream>>>(faces, xs, ys, zs,
                                                              coef, Nf, Fpad);
    // 256 tiles x 32 face-slices, one wave each
    dim3 grid(256, 32);
    raster_kernel<<<grid, 32, 0, stream>>>(coef, depth, nBatch);
}